// LightweightConvolution_10153302688260
// MI455X (gfx1250) — compile-verified
//
#include <hip/hip_runtime.h>
#include <stdint.h>

// Problem constants (match reference setup_inputs)
#define B_SZ   16
#define C_SZ   1024
#define H_SZ   16
#define KW     7
#define PADL   3
#define NLAYER 6

// Tiling
#define CH       32                 // channels per workgroup (half a head -> one shared kernel)
#define HALO     18                 // 6 layers * 3
#define LROWS    248                // LDS time rows per tile (incl. halo)
#define TT       (LROWS - 2*HALO)   // 212 output timesteps per tile
#define NTHREADS 256

__global__ void zero_out_kernel(float* __restrict__ o, int n) {
    int i = blockIdx.x * blockDim.x + threadIdx.x;
    if (i < n) o[i] = 0.0f;
}

__global__ __launch_bounds__(NTHREADS)
void lconv_fused_kernel(const float* __restrict__ x,
                        const float* __restrict__ w1, const float* __restrict__ w2,
                        const float* __restrict__ w3, const float* __restrict__ w4,
                        const float* __restrict__ w5, const float* __restrict__ w6,
                        float* __restrict__ out, int T)
{
    __shared__ float bufA[LROWS * CH];
    __shared__ float bufB[LROWS * CH];
    __shared__ float wsm[NLAYER][KW];

    const int tile   = blockIdx.x;
    const int cchunk = blockIdx.y;
    const int b      = blockIdx.z;
    const int tid    = threadIdx.x;
    const int c0     = cchunk * CH;
    const int head   = c0 / (C_SZ / H_SZ);
    const int g0     = tile * TT;          // first output timestep of this tile

    // ---- softmax-normalize the 6 per-head 7-tap kernels into LDS ----
    if (tid < NLAYER) {
        const float* w;
        switch (tid) {
            case 0:  w = w1; break;
            case 1:  w = w2; break;
            case 2:  w = w3; break;
            case 3:  w = w4; break;
            case 4:  w = w5; break;
            default: w = w6; break;
        }
        w += head * KW;
        float m = w[0];
        #pragma unroll
        for (int k = 1; k < KW; ++k) m = fmaxf(m, w[k]);
        float e[KW], s = 0.0f;
        #pragma unroll
        for (int k = 0; k < KW; ++k) { e[k] = __expf(w[k] - m); s += e[k]; }
        float inv = 1.0f / s;
        #pragma unroll
        for (int k = 0; k < KW; ++k) wsm[tid][k] = e[k] * inv;
    }

    // ---- zero-fill buffer A only when the tile touches the time boundary ----
    if (g0 - HALO < 0 || g0 - HALO + LROWS > T) {
        for (int i = tid; i < LROWS * CH; i += NTHREADS) bufA[i] = 0.0f;
    }
    __syncthreads();

    // ---- async global -> LDS tile load (CDNA5 ASYNCcnt path), 128b/lane ----
    {
        const int c4 = (tid & 7) * 4;     // float4 column within the 32-ch chunk
        const int r0 = tid >> 3;          // starting row, 32 rows loaded per pass
        for (int r = r0; r < LROWS; r += NTHREADS / 8) {
            int t = g0 - HALO + r;
            if (t >= 0 && t < T) {
                uint32_t lds = (uint32_t)(uintptr_t)(&bufA[r * CH + c4]);
                const float* g = x + ((size_t)t * (size_t)(B_SZ * C_SZ)
                                      + (size_t)b * C_SZ + c0 + c4);
                asm volatile("global_load_async_to_lds_b128 %0, %1, off"
                             :: "v"(lds), "v"((uint64_t)(uintptr_t)g)
                             : "memory");
            }
        }
    }
    asm volatile("s_wait_asynccnt 0" ::: "memory");
    __syncthreads();

    // ---- six fused conv layers in LDS (ping-pong, shrinking valid range) ----
    float* inb  = bufA;
    float* outb = bufB;
    const int c   = tid & (CH - 1);
    const int grp = tid >> 5;             // 0..7 row-groups (8 waves of rows)

    #pragma unroll 1
    for (int l = 1; l <= NLAYER; ++l) {
        float wl[KW];
        #pragma unroll
        for (int k = 0; k < KW; ++k) wl[k] = wsm[l - 1][k];
        const bool has_res = ((l & 1) == 0);

        const int lo   = 3 * l;
        const int hi   = LROWS - 3 * l;
        const int span = hi - lo;
        const int per  = (span + 7) >> 3;
        const int rs   = lo + grp * per;
        const int re   = min(rs + per, hi);

        if (rs < re) {
            // 7-tap sliding window in registers: 1 LDS read per output
            float win[KW];
            #pragma unroll
            for (int i = 0; i < KW - 1; ++i)
                win[i] = inb[(rs - PADL + i) * CH + c];
            for (int r = rs; r < re; ++r) {
                win[KW - 1] = inb[(r + PADL) * CH + c];
                float v = 0.0f;
                #pragma unroll
                for (int k = 0; k < KW; ++k) v = fmaf(wl[k], win[k], v);
                if (has_res) v += outb[r * CH + c];   // in-place residual (same thread r/w)
                v = fmaxf(v, 0.0f);
                int t = g0 - HALO + r;
                if (t < 0 || t >= T) v = 0.0f;        // keep zero-padding invariant
                outb[r * CH + c] = v;
                #pragma unroll
                for (int i = 0; i < KW - 1; ++i) win[i] = win[i + 1];
            }
        }
        __syncthreads();
        float* tmp = inb; inb = outb; outb = tmp;     // after l=6, inb holds the result
    }

    // ---- mean over time: partial sums per channel, then one atomic/channel ----
    const int nvalid = min(TT, T - g0);
    {
        float s = 0.0f;
        for (int r = HALO + grp; r < HALO + nvalid; r += 8)
            s += inb[r * CH + c];
        outb[grp * CH + c] = s;                       // outb (bufB) is free scratch now
    }
    __syncthreads();
    if (tid < CH) {
        float s = 0.0f;
        #pragma unroll
        for (int g = 0; g < 8; ++g) s += outb[g * CH + tid];
        unsafeAtomicAdd(&out[b * C_SZ + c0 + tid], s * (1.0f / (float)T));
    }
}

extern "C" void kernel_launch(void* const* d_in, const int* in_sizes, int n_in,
                              void* d_out, int out_size, void* d_ws, size_t ws_size,
                              hipStream_t stream) {
    const float* x  = (const float*)d_in[0];
    const float* w1 = (const float*)d_in[1];
    const float* w2 = (const float*)d_in[2];
    const float* w3 = (const float*)d_in[3];
    const float* w4 = (const float*)d_in[4];
    const float* w5 = (const float*)d_in[5];
    const float* w6 = (const float*)d_in[6];
    float* out = (float*)d_out;

    const int T  = in_sizes[0] / (B_SZ * C_SZ);      // 2048
    const int nt = (T + TT - 1) / TT;                // time tiles

    zero_out_kernel<<<(out_size + 255) / 256, 256, 0, stream>>>(out, out_size);

    dim3 grid(nt, C_SZ / CH, B_SZ);
    lconv_fused_kernel<<<grid, NTHREADS, 0, stream>>>(x, w1, w2, w3, w4, w5, w6, out, T);
}